// EfficientDetWithPostProcessing_43336220016741
// MI455X (gfx1250) — compile-verified
//
#include <hip/hip_runtime.h>
#include <hip/hip_bf16.h>
#include <math.h>

// Problem constants (from reference)
#define A_CNT   49104
#define A_PAD   49152          // multiple of 4096 (=> 12 float4 chunks per thread)
#define CLS     90
#define MAX_DET 100
#define IMG_F   512.0f
#define TILE_ANCHORS 256
#define WAVE    32             // gfx1250 is wave32-only

// ---------------------------------------------------------------------------
// Async copy helper: one 16-byte chunk, global -> LDS (ASYNCcnt-tracked).
// GVS mode: mem = SADDR(64b) + VADDR(32b offset); LDS dest = value in vdst VGPR.
// The single __shared__ object in each kernel sits at LDS offset 0, so the
// integer LDS address equals the byte offset into it.
// ---------------------------------------------------------------------------
__device__ __forceinline__ void async_g2l_b128(unsigned lds_off, unsigned mem_off,
                                               unsigned long long base) {
    asm volatile("global_load_async_to_lds_b128 %0, %1, %2"
                 :
                 : "v"(lds_off), "v"(mem_off), "s"(base)
                 : "memory");
}

__device__ __forceinline__ void wait_async_zero() {
    asm volatile("s_wait_asynccnt 0" ::: "memory");
}

// first-occurrence argmax combine rule (matches jnp.argmax tie semantics)
__device__ __forceinline__ void combine_max(float& bv, int& bi, float ov, int oi) {
    if (ov > bv || (ov == bv && oi < bi)) { bv = ov; bi = oi; }
}

// ---------------------------------------------------------------------------
// Kernel A: per-anchor class max/argmax (from LDS-staged classification tile)
//           + box decode + clip.  Batch 0 only (reference returns index [0]).
// ---------------------------------------------------------------------------
__global__ __launch_bounds__(TILE_ANCHORS)
void decode_score_kernel(const float* __restrict__ regression,      // [A,4] batch 0
                         const float* __restrict__ classification,  // [A,90] batch 0
                         const float* __restrict__ anchors,         // [A,4]
                         float4* __restrict__ boxes_ws,             // [A]
                         float*  __restrict__ scores_ws,            // [A_PAD]
                         int*    __restrict__ cls_ws)               // [A]
{
    __shared__ float tile[TILE_ANCHORS * CLS];   // 90 KB, LDS offset 0

    const int tid = threadIdx.x;
    const int a0  = blockIdx.x * TILE_ANCHORS;

    // Stage 256 contiguous classification rows (256*90*4 = 92160 B = 5760 chunks).
    // Last block reads a few rows past batch 0 — still inside the 8-batch buffer.
    const unsigned long long base =
        (unsigned long long)(const void*)(classification + (size_t)a0 * CLS);
    const int chunks = TILE_ANCHORS * CLS * 4 / 16;   // 5760
    for (int i = tid; i < chunks; i += TILE_ANCHORS) {
        unsigned off = (unsigned)(i * 16);
        async_g2l_b128(off, off, base);
    }
    wait_async_zero();
    __syncthreads();

    const int a = a0 + tid;
    if (a < A_CNT) {
        // max / argmax over 90 classes (strict > keeps first index)
        const float* row = &tile[tid * CLS];
        float best = -1e30f;
        int   bc   = 0;
        #pragma unroll 6
        for (int c = 0; c < CLS; ++c) {
            float v = row[c];
            if (v > best) { best = v; bc = c; }
        }

        // decode: anchors stored [y1,x1,y2,x2]; regression [dy,dx,dh,dw]
        float4 rg = ((const float4*)regression)[a];
        float4 an = ((const float4*)anchors)[a];
        float ya = (an.x + an.z) * 0.5f;
        float xa = (an.y + an.w) * 0.5f;
        float ha = an.z - an.x;
        float wa = an.w - an.y;
        float w  = expf(rg.w) * wa;
        float h  = expf(rg.z) * ha;
        float yc = rg.x * ha + ya;
        float xc = rg.y * wa + xa;
        float x1 = fmaxf(xc - w * 0.5f, 0.0f);
        float y1 = fmaxf(yc - h * 0.5f, 0.0f);
        float x2 = fminf(xc + w * 0.5f, IMG_F);
        float y2 = fminf(yc + h * 0.5f, IMG_F);

        boxes_ws[a]  = make_float4(x1, y1, x2, y2);
        scores_ws[a] = (best > 0.5f) ? best : -1.0f;   // masked score
        cls_ws[a]    = bc;
    } else if (a < A_PAD) {
        scores_ws[a] = -2.0f;   // pad: can never win argmax, deterministic
    }
}

// ---------------------------------------------------------------------------
// Kernel B: 100-round NMS. Whole score array lives in LDS (CDNA5: 320 KB/WGP).
// Single block, 1024 threads (32 waves). float4 LDS scans + wave32 shuffle
// argmax: 3 barriers per round instead of 12.
// ---------------------------------------------------------------------------
__global__ __launch_bounds__(1024)
void nms_kernel(const float4* __restrict__ boxes,     // [A]
                const float*  __restrict__ scores_g,  // [A_PAD]
                const int*    __restrict__ cls,       // [A]
                float* __restrict__ out)              // 400 boxes + 100 scores + 100 cls
{
    struct __align__(16) Smem {
        float scores[A_PAD];   // 192 KB, LDS offset 0, 16B aligned
        float red_v[WAVE];
        int   red_i[WAVE];
        float bbox[4];
        float barea;
    };
    __shared__ Smem sm;

    const int tid     = threadIdx.x;
    const int lane    = tid & (WAVE - 1);
    const int wave_id = tid >> 5;

    // Async-stage the masked scores into LDS: 49152*4 B = 12288 b128 chunks.
    const unsigned long long base = (unsigned long long)(const void*)scores_g;
    for (int i = tid; i < A_PAD / 4; i += 1024) {
        unsigned off = (unsigned)(i * 16);
        async_g2l_b128(off, off, base);   // sm.scores is at LDS offset 0
    }
    wait_async_zero();
    __syncthreads();

    const float4* s4 = (const float4*)sm.scores;

    for (int it = 0; it < MAX_DET; ++it) {
        // ---- per-thread stripe argmax, vectorized ds_load_b128 ----
        float bv = -1e30f;
        int   bi = 0x7fffffff;
        #pragma unroll
        for (int k = 0; k < (A_PAD / 4) / 1024; ++k) {     // 12 iterations
            int j = tid + (k << 10);
            float4 v = s4[j];
            int e = j << 2;
            if (v.x > bv) { bv = v.x; bi = e;     }
            if (v.y > bv) { bv = v.y; bi = e + 1; }
            if (v.z > bv) { bv = v.z; bi = e + 2; }
            if (v.w > bv) { bv = v.w; bi = e + 3; }
        }

        // ---- wave32 shuffle reduce (first-index tie rule) ----
        #pragma unroll
        for (int m = WAVE / 2; m > 0; m >>= 1) {
            float ov = __shfl_xor(bv, m, WAVE);
            int   oi = __shfl_xor(bi, m, WAVE);
            combine_max(bv, bi, ov, oi);
        }
        if (lane == 0) { sm.red_v[wave_id] = bv; sm.red_i[wave_id] = bi; }
        __syncthreads();                                    // barrier 1

        // ---- final reduce by wave 0, bookkeeping by lane 0 ----
        if (tid < WAVE) {
            float wv = sm.red_v[lane];
            int   wi = sm.red_i[lane];
            #pragma unroll
            for (int m = WAVE / 2; m > 0; m >>= 1) {
                float ov = __shfl_xor(wv, m, WAVE);
                int   oi = __shfl_xor(wi, m, WAVE);
                combine_max(wv, wi, ov, oi);
            }
            if (tid == 0) {
                float4 bb = boxes[wi];
                sm.bbox[0] = bb.x; sm.bbox[1] = bb.y;
                sm.bbox[2] = bb.z; sm.bbox[3] = bb.w;
                sm.barea   = (bb.z - bb.x) * (bb.w - bb.y);
                sm.scores[wi] = -1.0f;          // reference: s.at[best].set(-1)
                if (wv > 0.0f) {
                    out[it * 4 + 0]       = bb.x;
                    out[it * 4 + 1]       = bb.y;
                    out[it * 4 + 2]       = bb.z;
                    out[it * 4 + 3]       = bb.w;
                    out[4 * MAX_DET + it] = wv;             // masked==unmasked when >0.5
                    out[5 * MAX_DET + it] = (float)cls[wi];
                } else {
                    out[it * 4 + 0] = 0.0f; out[it * 4 + 1] = 0.0f;
                    out[it * 4 + 2] = 0.0f; out[it * 4 + 3] = 0.0f;
                    out[4 * MAX_DET + it] = 0.0f;
                    out[5 * MAX_DET + it] = 0.0f;
                }
            }
        }
        __syncthreads();                                    // barrier 2

        // ---- unconditional IoU suppression against best box ----
        const float bx1 = sm.bbox[0], by1 = sm.bbox[1];
        const float bx2 = sm.bbox[2], by2 = sm.bbox[3];
        const float ba  = sm.barea;
        for (int a = tid; a < A_CNT; a += 1024) {
            if (a + 1024 < A_CNT)
                __builtin_prefetch(&boxes[a + 1024], 0, 1); // global_prefetch_b8
            float4 cb = boxes[a];
            float xx1 = fmaxf(bx1, cb.x);
            float yy1 = fmaxf(by1, cb.y);
            float xx2 = fminf(bx2, cb.z);
            float yy2 = fminf(by2, cb.w);
            float inter = fmaxf(xx2 - xx1, 0.0f) * fmaxf(yy2 - yy1, 0.0f);
            float ca    = (cb.z - cb.x) * (cb.w - cb.y);
            float iou   = inter / (ba + ca - inter + 1e-8f);
            if (iou > 0.5f) sm.scores[a] = -1.0f;
        }
        __syncthreads();                                    // barrier 3
    }
}

// ---------------------------------------------------------------------------
// Host-side launch. Inputs (setup_inputs order): x, regression, classification,
// anchors. Output: boxes[100,4] ++ scores[100] ++ classes[100] (600 floats).
// ---------------------------------------------------------------------------
extern "C" void kernel_launch(void* const* d_in, const int* in_sizes, int n_in,
                              void* d_out, int out_size, void* d_ws, size_t ws_size,
                              hipStream_t stream) {
    (void)in_sizes; (void)n_in; (void)out_size; (void)ws_size;

    const float* regression     = (const float*)d_in[1];   // [8, A, 4]  -> batch 0
    const float* classification = (const float*)d_in[2];   // [8, A, 90] -> batch 0
    const float* anchors        = (const float*)d_in[3];   // [A, 4]
    float* out = (float*)d_out;

    // Workspace layout (bytes): scores[A_PAD] | boxes[A] float4 | cls[A] int
    char* ws = (char*)d_ws;
    float*  scores_ws = (float*)ws;                                    // 196608 B
    float4* boxes_ws  = (float4*)(ws + (size_t)A_PAD * sizeof(float)); // 785664 B
    int*    cls_ws    = (int*)(ws + (size_t)A_PAD * sizeof(float)
                                  + (size_t)A_CNT * sizeof(float4));   // 196416 B

    const int gridA = (A_CNT + TILE_ANCHORS - 1) / TILE_ANCHORS;  // 192
    decode_score_kernel<<<gridA, TILE_ANCHORS, 0, stream>>>(
        regression, classification, anchors, boxes_ws, scores_ws, cls_ws);

    nms_kernel<<<1, 1024, 0, stream>>>(boxes_ws, scores_ws, cls_ws, out);
}